// GeneralConv_43370579755358
// MI455X (gfx1250) — compile-verified
//
#include <hip/hip_runtime.h>
#include <math.h>

#define N_NODES 50000
#define N_EDGES 800000
#define DIM     128
#define NTYPE   4
#define NHEAD   8
#define HDF     8          // h feature dim
#define NPAD    (N_NODES + NTYPE*16)   // padded permuted-node list length

typedef __attribute__((ext_vector_type(2))) float v2f;
typedef __attribute__((ext_vector_type(8))) float v8f;
typedef __attribute__((address_space(3))) char lds_char;

// ---------------- small utility kernels ----------------

__global__ void fill_u32_kernel(unsigned* __restrict__ p, unsigned v, int n) {
    int i = blockIdx.x * blockDim.x + threadIdx.x;
    if (i < n) p[i] = v;
}

__global__ void hist_kernel(const int* __restrict__ ntype, int* __restrict__ cnt) {
    int i = blockIdx.x * blockDim.x + threadIdx.x;
    if (i < N_NODES) atomicAdd(&cnt[ntype[i]], 1);
}

__global__ void prefix_kernel(const int* __restrict__ cnt, int* __restrict__ start) {
    if (blockIdx.x == 0 && threadIdx.x == 0) {
        int off = 0;
        for (int t = 0; t < NTYPE; ++t) {
            start[t] = off;
            off += (cnt[t] + 15) & ~15;   // 16-align each type bucket
        }
    }
}

__global__ void scatter_kernel(const int* __restrict__ ntype, const int* __restrict__ start,
                               int* __restrict__ cursor, int* __restrict__ perm) {
    int i = blockIdx.x * blockDim.x + threadIdx.x;
    if (i < N_NODES) {
        int t = ntype[i];
        int pos = start[t] + atomicAdd(&cursor[t], 1);
        perm[pos] = i;
    }
}

// ---------------- h-projections: [N,8] @ [8,64] x2 ----------------

__global__ void hproj_kernel(const float* __restrict__ h_mat,
                             const float* __restrict__ hsub_W, const float* __restrict__ hsub_b,
                             const float* __restrict__ hneigh_W, const float* __restrict__ hneigh_b,
                             float* __restrict__ hsub, float* __restrict__ hneigh) {
    int gid = blockIdx.x * blockDim.x + threadIdx.x;
    if (gid >= N_NODES * 64) return;
    int n = gid >> 6;
    int o = gid & 63;
    float as = hsub_b[o];
    float an = hneigh_b[o];
#pragma unroll
    for (int k = 0; k < HDF; ++k) {
        float hv = h_mat[n * HDF + k];
        as += hv * hsub_W[k * 64 + o];
        an += hv * hneigh_W[k * 64 + o];
    }
    hsub[gid]   = as;
    hneigh[gid] = an;
}

// ---------------- type-bucketed q/k GEMM with fp32 WMMA ----------------
// One workgroup = 16-row tile of the permuted node list; 8 waves, one 16-col
// tile each; K=128 swept with 32x V_WMMA_F32_16X16X4_F32 per output matrix.
// X tile staged directly into LDS with per-lane async loads (ASYNCcnt path).

__global__ __launch_bounds__(256) void qk_gemm_kernel(
    const float* __restrict__ X, const int* __restrict__ ntype,
    const float* __restrict__ subW, const float* __restrict__ subB,
    const float* __restrict__ neighW, const float* __restrict__ neighB,
    const int* __restrict__ perm,
    float* __restrict__ qout, float* __restrict__ kout)
{
    __shared__ float Xs[16][132];   // 132-float row stride keeps 16B alignment, dodges bank conflicts
    __shared__ int nid[16];
    __shared__ int ttypeS;

    const int tid  = threadIdx.x;
    const int base = blockIdx.x * 16;

    if (tid < 16) nid[tid] = perm[base + tid];
    __syncthreads();
    if (tid == 0) {
        int tt = -1;
        for (int r = 0; r < 16; ++r) {
            int n = nid[r];
            if (n >= 0) { tt = ntype[n]; break; }
        }
        ttypeS = tt;
    }
    __syncthreads();
    const int tt = ttypeS;
    if (tt < 0) return;          // fully-padded tile (uniform exit: EXEC all-1s at WMMA)

    // Stage X tile: 16 rows x 128 cols = 512 float4, 2 per thread.
    // Per-lane async global->LDS copy (GLOBAL_LOAD_ASYNC_TO_LDS_B128):
    // bypasses VGPRs entirely. Padded rows (-1) fetch node 0; their outputs
    // are never stored, and WMMA rows are independent, so this is safe.
    for (int j = tid; j < 512; j += 256) {
        int r  = j >> 5;
        int c4 = j & 31;
        int n  = nid[r];
        int nn = (n >= 0) ? n : 0;
        const float* gp = X + (size_t)nn * DIM + c4 * 4;
        lds_char* lp = (lds_char*)&Xs[r][c4 * 4];
        asm volatile("global_load_async_to_lds_b128 %0, %1, off"
                     :: "v"(lp), "v"(gp) : "memory");
    }
    asm volatile("s_wait_asynccnt 0x0" ::: "memory");
    __syncthreads();

    const int wave = tid >> 5;
    const int lane = tid & 31;
    const int row  = lane & 15;            // A: lanes 0-15/16-31 both hold M=0..15
    const int koff = (lane >> 4) * 2;      // A/B: upper lane half holds K+2..K+3
    const int col  = wave * 16 + (lane & 15);

    const float* __restrict__ Wq = subW   + (size_t)tt * DIM * DIM;
    const float* __restrict__ Wk = neighW + (size_t)tt * DIM * DIM;

    // Warm the L2/L0 path for the weight panels (global_prefetch_b8)
#pragma unroll
    for (int p = 0; p < 4; ++p) {
        __builtin_prefetch(Wq + (p * 32 + lane) * DIM + col, 0, 0);
        __builtin_prefetch(Wk + (p * 32 + lane) * DIM + col, 0, 0);
    }

    v8f accq = {};
    v8f acck = {};
#pragma unroll
    for (int kk = 0; kk < 32; ++kk) {
        const int kb = kk * 4 + koff;
        v2f a, bq, bk;
        a.x  = Xs[row][kb];
        a.y  = Xs[row][kb + 1];
        bq.x = Wq[kb * DIM + col];
        bq.y = Wq[(kb + 1) * DIM + col];
        bk.x = Wk[kb * DIM + col];
        bk.y = Wk[(kb + 1) * DIM + col];
        accq = __builtin_amdgcn_wmma_f32_16x16x4_f32(false, a, false, bq, (short)0, accq, false, false);
        acck = __builtin_amdgcn_wmma_f32_16x16x4_f32(false, a, false, bk, (short)0, acck, false, false);
    }

    const float biasq = subB[tt * DIM + col];
    const float biask = neighB[tt * DIM + col];
    const int mbase = (lane >> 4) * 8;     // C/D: upper lane half holds M+8
#pragma unroll
    for (int v = 0; v < 8; ++v) {
        int n = nid[mbase + v];
        if (n >= 0) {
            qout[(size_t)n * DIM + col] = accq[v] + biasq;
            kout[(size_t)n * DIM + col] = acck[v] + biask;
        }
    }
}

// ---------------- edge attention ----------------

__device__ __forceinline__ unsigned fkey(float f) {
    unsigned u = __float_as_uint(f);
    return u ^ (unsigned)(((int)u >> 31) | (int)0x80000000);
}
__device__ __forceinline__ float fval(unsigned k) {
    unsigned u = (k & 0x80000000u) ? (k ^ 0x80000000u) : ~k;
    return __uint_as_float(u);
}

__global__ void edge_att_kernel(const int* __restrict__ ei, const int* __restrict__ ntype,
                                const float* __restrict__ hsub, const float* __restrict__ hneigh,
                                const float* __restrict__ qn, const float* __restrict__ kn,
                                const float* __restrict__ ratt, const float* __restrict__ rhatt,
                                float* __restrict__ att, unsigned* __restrict__ mkey) {
    int gid = blockIdx.x * blockDim.x + threadIdx.x;
    if (gid >= N_EDGES * NHEAD) return;
    int e = gid >> 3;
    int h = gid & 7;
    int src = ei[e];
    int dst = ei[N_EDGES + e];
    int st  = ntype[dst];

    const float* hs = hsub   + (size_t)dst * 64 + h * 8;
    const float* hn = hneigh + (size_t)src * 64 + h * 8;
    const float* rh = rhatt  + st * 64 + h * 8;
    float ah = 0.f;
#pragma unroll
    for (int k = 0; k < 8; ++k) ah += tanhf(hs[k] * hn[k]) * rh[k];
    ah *= 0.35355339059327373f;            // 1/sqrt(H)

    const float* qp = qn  + (size_t)dst * DIM + h * 16;
    const float* kp = kn  + (size_t)src * DIM + h * 16;
    const float* ra = ratt + st * 128 + h * 16;
    float s = 0.f;
#pragma unroll
    for (int k = 0; k < 16; ++k) s += tanhf(qp[k] * kp[k]) * ra[k];

    float a = s * 0.25f * ah;              // /sqrt(DK) * att_h
    att[gid] = a;
    atomicMax(&mkey[(size_t)dst * NHEAD + h], fkey(a));
}

__global__ void edge_exp_kernel(const int* __restrict__ ei, const unsigned* __restrict__ mkey,
                                float* __restrict__ att, float* __restrict__ denom) {
    int gid = blockIdx.x * blockDim.x + threadIdx.x;
    if (gid >= N_EDGES * NHEAD) return;
    int e = gid >> 3;
    int h = gid & 7;
    int dst = ei[N_EDGES + e];
    float m  = fval(mkey[(size_t)dst * NHEAD + h]);
    float ea = expf(att[gid] - m);
    att[gid] = ea;
    atomicAdd(&denom[(size_t)dst * NHEAD + h], ea);
}

__global__ void edge_aggr_kernel(const int* __restrict__ ei, const float* __restrict__ att,
                                 const float* __restrict__ denom, const float* __restrict__ kn,
                                 float* __restrict__ aggr) {
    int gid = blockIdx.x * blockDim.x + threadIdx.x;
    if (gid >= N_EDGES * DIM) return;
    int e = gid >> 7;
    int d = gid & 127;
    int src = ei[e];
    int dst = ei[N_EDGES + e];
    int h = d >> 4;
    float alpha = att[(size_t)e * NHEAD + h] / (denom[(size_t)dst * NHEAD + h] + 1e-16f);
    atomicAdd(&aggr[(size_t)dst * DIM + d], kn[(size_t)src * DIM + d] * alpha);
}

// ---------------- GELU + LayerNorm, one wave32 per node ----------------

__device__ __forceinline__ float gelu_exact(float x) {
    return 0.5f * x * (1.0f + erff(x * 0.70710678118654752f));
}

__global__ __launch_bounds__(256) void finalize_kernel(const float* __restrict__ aggr,
                                                       const float* __restrict__ gamma,
                                                       const float* __restrict__ beta,
                                                       float* __restrict__ out) {
    int lane = threadIdx.x & 31;
    int node = blockIdx.x * 8 + (threadIdx.x >> 5);
    if (node >= N_NODES) return;

    float4 x = ((const float4*)(aggr + (size_t)node * DIM))[lane];
    float g0 = gelu_exact(x.x), g1 = gelu_exact(x.y), g2 = gelu_exact(x.z), g3 = gelu_exact(x.w);
    float s  = g0 + g1 + g2 + g3;
    float ss = g0 * g0 + g1 * g1 + g2 * g2 + g3 * g3;
#pragma unroll
    for (int off = 16; off >= 1; off >>= 1) {
        s  += __shfl_xor(s, off);
        ss += __shfl_xor(ss, off);
    }
    float mean = s * (1.0f / 128.0f);
    float var  = ss * (1.0f / 128.0f) - mean * mean;
    float rs   = rsqrtf(var + 1e-5f);
    int c = lane * 4;
    float4 o;
    o.x = (g0 - mean) * rs * gamma[c + 0] + beta[c + 0];
    o.y = (g1 - mean) * rs * gamma[c + 1] + beta[c + 1];
    o.z = (g2 - mean) * rs * gamma[c + 2] + beta[c + 2];
    o.w = (g3 - mean) * rs * gamma[c + 3] + beta[c + 3];
    ((float4*)(out + (size_t)node * DIM))[lane] = o;
}

// ---------------- host launcher ----------------

extern "C" void kernel_launch(void* const* d_in, const int* in_sizes, int n_in,
                              void* d_out, int out_size, void* d_ws, size_t ws_size,
                              hipStream_t stream) {
    const float* meta_xs   = (const float*)d_in[0];
    const int*   node_type = (const int*)  d_in[1];
    const int*   edge_idx  = (const int*)  d_in[2];
    const float* h_mat     = (const float*)d_in[3];
    const float* sub_W     = (const float*)d_in[4];
    const float* sub_b     = (const float*)d_in[5];
    const float* neigh_W   = (const float*)d_in[6];
    const float* neigh_b   = (const float*)d_in[7];
    const float* hsub_W    = (const float*)d_in[8];
    const float* hsub_b    = (const float*)d_in[9];
    const float* hneigh_W  = (const float*)d_in[10];
    const float* hneigh_b  = (const float*)d_in[11];
    const float* rel_att   = (const float*)d_in[12];
    const float* rel_h_att = (const float*)d_in[13];
    const float* ln_gamma  = (const float*)d_in[14];
    const float* ln_beta   = (const float*)d_in[15];
    float* out = (float*)d_out;

    // workspace layout (32-bit words); zero-init region first, then perm (-1 fill)
    unsigned* ws = (unsigned*)d_ws;
    size_t off = 0;
    unsigned* mkey   = ws + off;            off += (size_t)N_NODES * NHEAD;
    float*    denom  = (float*)(ws + off);  off += (size_t)N_NODES * NHEAD;
    float*    aggr   = (float*)(ws + off);  off += (size_t)N_NODES * DIM;
    int*      cnt    = (int*)(ws + off);    off += NTYPE;
    int*      cursor = (int*)(ws + off);    off += NTYPE;
    const int zwords = (int)off;            // everything above must start at 0
    int*      perm   = (int*)(ws + off);    off += NPAD;
    int*      startb = (int*)(ws + off);    off += NTYPE;
    float*    qn     = (float*)(ws + off);  off += (size_t)N_NODES * DIM;
    float*    kn     = (float*)(ws + off);  off += (size_t)N_NODES * DIM;
    float*    hsub   = (float*)(ws + off);  off += (size_t)N_NODES * 64;
    float*    hneigh = (float*)(ws + off);  off += (size_t)N_NODES * 64;
    float*    attb   = (float*)(ws + off);  off += (size_t)N_EDGES * NHEAD;
    (void)ws_size; (void)in_sizes; (void)n_in; (void)out_size;

    fill_u32_kernel<<<(zwords + 255) / 256, 256, 0, stream>>>((unsigned*)d_ws, 0u, zwords);
    fill_u32_kernel<<<(NPAD + 255) / 256, 256, 0, stream>>>((unsigned*)perm, 0xFFFFFFFFu, NPAD);

    hist_kernel<<<(N_NODES + 255) / 256, 256, 0, stream>>>(node_type, cnt);
    prefix_kernel<<<1, 32, 0, stream>>>(cnt, startb);
    scatter_kernel<<<(N_NODES + 255) / 256, 256, 0, stream>>>(node_type, startb, cursor, perm);

    hproj_kernel<<<(N_NODES * 64 + 255) / 256, 256, 0, stream>>>(
        h_mat, hsub_W, hsub_b, hneigh_W, hneigh_b, hsub, hneigh);

    qk_gemm_kernel<<<NPAD / 16, 256, 0, stream>>>(
        meta_xs, node_type, sub_W, sub_b, neigh_W, neigh_b, perm, qn, kn);

    const int EH = N_EDGES * NHEAD;
    edge_att_kernel<<<(EH + 255) / 256, 256, 0, stream>>>(
        edge_idx, node_type, hsub, hneigh, qn, kn, rel_att, rel_h_att, attb, mkey);
    edge_exp_kernel<<<(EH + 255) / 256, 256, 0, stream>>>(edge_idx, mkey, attb, denom);
    edge_aggr_kernel<<<(N_EDGES * DIM + 255) / 256, 256, 0, stream>>>(
        edge_idx, attb, denom, kn, aggr);

    finalize_kernel<<<(N_NODES + 7) / 8, 256, 0, stream>>>(aggr, ln_gamma, ln_beta, out);
}